// FactorizationMachine_20358144983118
// MI455X (gfx1250) — compile-verified
//
#include <hip/hip_runtime.h>

// Problem constants (from reference): B=16384, N=2048, F=64
#define B_ROWS   16384
#define N_DIM    2048
#define F_DIM    64
#define KT       64          // K tile per iteration
#define NKT      (N_DIM / KT)
#define LDS_STR  68          // x tile row stride: 64 + 4 pad (bank-conflict-free A reads)
#define V_PSTR   160         // v tile k-pair row stride (floats): 128 data + 32 pad
                             // 160 % 64 == 32 -> the two lane-halves hit disjoint banks

typedef __attribute__((ext_vector_type(2))) float v2f;
typedef __attribute__((ext_vector_type(4))) float v4f;
typedef __attribute__((ext_vector_type(8))) float v8f;

#if __has_builtin(__builtin_amdgcn_global_load_async_to_lds_b32)
#define HAVE_ASYNC_LDS 1
#else
#define HAVE_ASYNC_LDS 0
#endif

__device__ __forceinline__ void wait_async0() {
#if __has_builtin(__builtin_amdgcn_s_wait_asynccnt)
    __builtin_amdgcn_s_wait_asynccnt(0);
#elif HAVE_ASYNC_LDS
    asm volatile("s_wait_asynccnt 0" ::: "memory");
#endif
}

// Stage one v k-slice [64 rows x 64 cols] into LDS, k-pair interleaved:
// element (k,f) -> buf[(k>>1)*V_PSTR + 2*f + (k&1)], so a WMMA B fragment
// (v[k][f], v[k+1][f]) is one contiguous, 8B-aligned b64 in LDS.
// 256 threads, flat global index g = tid + 256*i (coalesced b32 async loads).
__device__ __forceinline__ void stage_v_tile(const float* __restrict__ v, int k0,
                                             float* __restrict__ buf, int tid) {
    const int k = tid >> 6;          // 0..3 within first group of 4 rows
    const int f = tid & 63;
    const float* src = v + (size_t)k0 * F_DIM + tid;               // = v[(k0+k)][f]
    float* dst = buf + (k >> 1) * V_PSTR + f * 2 + (k & 1);
#pragma unroll
    for (int i = 0; i < 16; ++i) {
#if HAVE_ASYNC_LDS
        __builtin_amdgcn_global_load_async_to_lds_b32(
            (__attribute__((address_space(1))) int*)(src),
            (__attribute__((address_space(3))) int*)(dst),
            0, 0);
#else
        *dst = *src;
#endif
        src += 4 * F_DIM;            // +4 k rows in global
        dst += 2 * V_PSTR;           // +2 k-pairs in LDS
    }
}

// ---------------------------------------------------------------------------
// Pass 1: s[n] = sum_f v[n][f]^2
// ---------------------------------------------------------------------------
__global__ __launch_bounds__(256) void fm_precompute_s(const float* __restrict__ v,
                                                       float* __restrict__ s) {
    int n = blockIdx.x * blockDim.x + threadIdx.x;
    if (n >= N_DIM) return;
    const float* row = v + (size_t)n * F_DIM;
    float acc = 0.f;
#pragma unroll
    for (int f = 0; f < F_DIM; f += 4) {
        v4f t = *(const v4f*)(row + f);
        acc = fmaf(t.x, t.x, acc);
        acc = fmaf(t.y, t.y, acc);
        acc = fmaf(t.z, t.z, acc);
        acc = fmaf(t.w, t.w, acc);
    }
    s[n] = acc;
}

// ---------------------------------------------------------------------------
// Pass 2: fused FM forward.
// ---------------------------------------------------------------------------
__global__ __launch_bounds__(256) void fm_main(const float* __restrict__ x,
                                               const float* __restrict__ w0,
                                               const float* __restrict__ w1,
                                               const float* __restrict__ v,
                                               const float* __restrict__ s,
                                               float* __restrict__ out) {
    __shared__ float ldsA[8 * 16 * LDS_STR];       // per-wave x tiles       (~34.8 KB)
    __shared__ float ldsV[2 * 32 * V_PSTR];        // double-buffered v tile  (~41.0 KB)
    __shared__ float ldsOut[8 * 16];

    const int tid    = threadIdx.x;
    const int lane   = tid & 31;
    const int wv     = tid >> 5;
    const int laneLo = lane & 15;
    const int laneHi = lane >> 4;
    const int rowBase = blockIdx.x * 128 + wv * 16;
    const int col4   = laneLo * 4;

    float* myA = ldsA + wv * (16 * LDS_STR);

    v8f acc0 = {}, acc1 = {}, acc2 = {}, acc3 = {};
    float lin[8], sq[8];
#pragma unroll
    for (int j = 0; j < 8; ++j) { lin[j] = 0.f; sq[j] = 0.f; }

    // prologue: stage v tile 0
    stage_v_tile(v, 0, ldsV, tid);

    for (int kt = 0; kt < NKT; ++kt) {
        const int k0 = kt * KT;

        wait_async0();        // own async stores for tile kt have landed in LDS
        __syncthreads();      // everyone's portion visible; everyone done with kt-1

        // kick off next tile's staging (overlaps with compute below)
        if (kt + 1 < NKT)
            stage_v_tile(v, (kt + 1) * KT, ldsV + ((kt + 1) & 1) * (32 * V_PSTR), tid);

        const float* cv = ldsV + (kt & 1) * (32 * V_PSTR);

        // per-tile w1 / s segments
        v4f wseg = *(const v4f*)(w1 + k0 + col4);
        v4f sseg = *(const v4f*)(s  + k0 + col4);

        // ---- stage x tile, fuse lin/sq partials
#pragma unroll
        for (int j = 0; j < 8; ++j) {
            const int row = 2 * j + laneHi;
            const float* p = x + (size_t)(rowBase + row) * N_DIM + k0 + col4;
            v4f xr = __builtin_nontemporal_load((const v4f*)p);

            lin[j] = fmaf(xr.x, wseg.x, lin[j]);
            lin[j] = fmaf(xr.y, wseg.y, lin[j]);
            lin[j] = fmaf(xr.z, wseg.z, lin[j]);
            lin[j] = fmaf(xr.w, wseg.w, lin[j]);
            sq[j]  = fmaf(xr.x * xr.x, sseg.x, sq[j]);
            sq[j]  = fmaf(xr.y * xr.y, sseg.y, sq[j]);
            sq[j]  = fmaf(xr.z * xr.z, sseg.z, sq[j]);
            sq[j]  = fmaf(xr.w * xr.w, sseg.w, sq[j]);

            *(v4f*)(myA + row * LDS_STR + col4) = xr;

            if (kt + 1 < NKT)
                __builtin_prefetch(p + KT, 0, 1);   // global_prefetch_b8
        }

        // ---- 16 WMMA K-steps of 4
#pragma unroll
        for (int ks = 0; ks < KT / 4; ++ks) {
            // A fragment: lane m=laneLo holds K pair (laneHi*2, laneHi*2+1)
            v2f a = *(const v2f*)(myA + laneLo * LDS_STR + ks * 4 + laneHi * 2);

            // B fragments: contiguous (v[k][f], v[k+1][f]) pairs, k = 4ks + 2*laneHi
            const float* vb = cv + (2 * ks + laneHi) * V_PSTR + laneLo * 2;
            v2f b0 = *(const v2f*)(vb + 0);    // f0 = 0
            v2f b1 = *(const v2f*)(vb + 32);   // f0 = 16
            v2f b2 = *(const v2f*)(vb + 64);   // f0 = 32
            v2f b3 = *(const v2f*)(vb + 96);   // f0 = 48

            acc0 = __builtin_amdgcn_wmma_f32_16x16x4_f32(false, a, false, b0,
                                                         (short)0, acc0, false, false);
            acc1 = __builtin_amdgcn_wmma_f32_16x16x4_f32(false, a, false, b1,
                                                         (short)0, acc1, false, false);
            acc2 = __builtin_amdgcn_wmma_f32_16x16x4_f32(false, a, false, b2,
                                                         (short)0, acc2, false, false);
            acc3 = __builtin_amdgcn_wmma_f32_16x16x4_f32(false, a, false, b3,
                                                         (short)0, acc3, false, false);
        }
    }

    // ---- epilogue: 0.5 * rowsum(xv^2). C layout: reg r, half h -> row M = r + 8h.
#pragma unroll
    for (int r = 0; r < 8; ++r) {
        float t = acc0[r] * acc0[r] + acc1[r] * acc1[r] +
                  acc2[r] * acc2[r] + acc3[r] * acc3[r];
        t += __shfl_xor(t, 1, 32);
        t += __shfl_xor(t, 2, 32);
        t += __shfl_xor(t, 4, 32);
        t += __shfl_xor(t, 8, 32);
        if (laneLo == 0)
            ldsOut[wv * 16 + laneHi * 8 + r] = 0.5f * t;
    }

    // ---- scalar terms: row 2j+laneHi partial across 16 lanes
#pragma unroll
    for (int j = 0; j < 8; ++j) {
        float t = lin[j] - 0.5f * sq[j];
        t += __shfl_xor(t, 1, 32);
        t += __shfl_xor(t, 2, 32);
        t += __shfl_xor(t, 4, 32);
        t += __shfl_xor(t, 8, 32);
        if (laneLo == 0)
            ldsOut[wv * 16 + 2 * j + laneHi] += t;   // same-wave DS ops are in-order
    }

    if (lane < 16)
        out[rowBase + lane] = w0[0] + ldsOut[wv * 16 + lane];
}

// ---------------------------------------------------------------------------
extern "C" void kernel_launch(void* const* d_in, const int* in_sizes, int n_in,
                              void* d_out, int out_size, void* d_ws, size_t ws_size,
                              hipStream_t stream) {
    const float* x  = (const float*)d_in[0];   // [16384, 2048]
    const float* w0 = (const float*)d_in[1];   // [1]
    const float* w1 = (const float*)d_in[2];   // [2048]
    const float* v  = (const float*)d_in[3];   // [2048, 64]
    float* s   = (float*)d_ws;                 // [2048] scratch
    float* out = (float*)d_out;                // [16384]

    fm_precompute_s<<<N_DIM / 256, 256, 0, stream>>>(v, s);
    fm_main<<<B_ROWS / 128, 256, 0, stream>>>(x, w0, w1, v, s, out);
}